// GraphSpectralFilter_60653528154335
// MI455X (gfx1250) — compile-verified
//
#include <hip/hip_runtime.h>
#include <hip/hip_bf16.h>

typedef __attribute__((ext_vector_type(2))) float v2f;
typedef __attribute__((ext_vector_type(8))) float v8f;

#define D 64  // D_IN == D_OUT == 64

// ---------------------------------------------------------------------------
// Phase 1: degree.  deg[i] = 1 (self loop) + sum of edge weights into i.
// ---------------------------------------------------------------------------
__global__ void k_deg_init(float* __restrict__ deg, int N) {
    int i = blockIdx.x * blockDim.x + threadIdx.x;
    if (i < N) deg[i] = 1.0f;  // self-loop weight
}

__global__ void k_deg_edges(const int* __restrict__ dst,
                            const float* __restrict__ ew,
                            float* __restrict__ deg, int E) {
    int e = blockIdx.x * blockDim.x + threadIdx.x;
    if (e < E) atomicAdd(&deg[dst[e]], ew[e]);
}

__global__ void k_dinv(const float* __restrict__ deg,
                       float* __restrict__ dinv, int N) {
    int i = blockIdx.x * blockDim.x + threadIdx.x;
    if (i < N) {
        float d = deg[i];
        dinv[i] = (d > 0.0f) ? rsqrtf(d) : 0.0f;
    }
}

// ---------------------------------------------------------------------------
// Phase 2: aggregation in x-space (aggregate-then-transform).
//   agg[i] = dinv[i]^2 * x[i]                      (self loop, norm = 1/deg)
//   agg[dst] += dinv[src]*ew*dinv[dst] * x[src]    (edges)
// ---------------------------------------------------------------------------
__global__ void k_agg_init(const float* __restrict__ x,
                           const float* __restrict__ dinv,
                           float* __restrict__ agg, int N) {
    int idx = blockIdx.x * blockDim.x + threadIdx.x;  // N*16 work items, 4 floats each
    if (idx >= N * (D / 4)) return;
    int i = idx >> 4;
    int j = (idx & 15) << 2;
    float s  = dinv[i];
    float s2 = s * s;
    float4 xv = *(const float4*)(x + i * D + j);
    float4 r;
    r.x = xv.x * s2; r.y = xv.y * s2; r.z = xv.z * s2; r.w = xv.w * s2;
    *(float4*)(agg + i * D + j) = r;
}

__global__ void k_scatter(const int* __restrict__ src,
                          const int* __restrict__ dst,
                          const float* __restrict__ ew,
                          const float* __restrict__ dinv,
                          const float* __restrict__ x,
                          float* __restrict__ agg, int E) {
    int t = blockIdx.x * blockDim.x + threadIdx.x;
    int e = t >> 5;          // one wave32 per edge
    int lane = t & 31;       // each lane handles 2 consecutive features
    if (e >= E) return;
    int s = src[e];
    int d = dst[e];
    float n = dinv[s] * ew[e] * dinv[d];
    float2 xv = ((const float2*)(x + s * D))[lane];
    atomicAdd(&agg[d * D + lane * 2 + 0], n * xv.x);
    atomicAdd(&agg[d * D + lane * 2 + 1], n * xv.y);
}

// ---------------------------------------------------------------------------
// Phase 3: fp32 WMMA GEMM.  out[n, k*64+j] = sum_d agg[n,d]*W[k,d,j] + b[k,j]
// One wave computes a 16x64 tile (rows = 16-row strip, cols = head k).
// K-dim 64 -> 16 chained V_WMMA_F32_16X16X4_F32 per 16x16 sub-tile.
//
// VGPR layouts (ISA 7.12.2, wave32):
//   A 16x4 f32:  lane L -> row M = L&15, K = (L>>4)*2 + {0,1} in {A.x, A.y}
//   B 4x16  f32: lane L -> col N = L&15, K = (L>>4)*2 + {0,1} in {B.x, B.y}
//   C/D 16x16:   vgpr r, lanes 0-15 -> (M=r, N=lane); lanes 16-31 -> (M=r+8)
// ---------------------------------------------------------------------------
__global__ __launch_bounds__(256) void k_gemm(const float* __restrict__ agg,
                                              const float* __restrict__ W,
                                              const float* __restrict__ bias,
                                              float* __restrict__ out,
                                              int N, int K) {
    const int lane = threadIdx.x & 31;
    const int k    = threadIdx.x >> 5;       // head index (0..K-1)
    const int rowblk = blockIdx.x;           // 16-row strip
    const int m  = lane & 15;                // row (A) / col (B,C)
    const int kk = (lane >> 4) << 1;         // K sub-offset {0,2}
    const int NC = K * D;                    // out row stride (512)

    int arow = rowblk * 16 + m;
    if (arow >= N) arow = N - 1;             // clamp (N is a multiple of 16 anyway)
    const float* Arow = agg + arow * D;
    const float* Wk   = W + k * D * D;       // W[k]: 64x64 row-major (d, j)

    v8f c0 = {}, c1 = {}, c2 = {}, c3 = {};
#pragma unroll
    for (int kb = 0; kb < D; kb += 4) {
        v2f a;
        a.x = Arow[kb + kk + 0];
        a.y = Arow[kb + kk + 1];
        const float* Wc = Wk + (kb + kk) * D + m;   // row kb+kk, col base m
        v2f b0 = { Wc[0],  Wc[D + 0]  };            // cols [ 0,16)
        v2f b1 = { Wc[16], Wc[D + 16] };            // cols [16,32)
        v2f b2 = { Wc[32], Wc[D + 32] };            // cols [32,48)
        v2f b3 = { Wc[48], Wc[D + 48] };            // cols [48,64)
        c0 = __builtin_amdgcn_wmma_f32_16x16x4_f32(false, a, false, b0, (short)0, c0, false, false);
        c1 = __builtin_amdgcn_wmma_f32_16x16x4_f32(false, a, false, b1, (short)0, c1, false, false);
        c2 = __builtin_amdgcn_wmma_f32_16x16x4_f32(false, a, false, b2, (short)0, c2, false, false);
        c3 = __builtin_amdgcn_wmma_f32_16x16x4_f32(false, a, false, b3, (short)0, c3, false, false);
    }

    const float* bk = bias + k * D;
    float bv0 = bk[m +  0];
    float bv1 = bk[m + 16];
    float bv2 = bk[m + 32];
    float bv3 = bk[m + 48];

    const int rowhalf = (lane >> 4) * 8;     // lanes 16-31 hold rows M+8
    int row0 = rowblk * 16 + rowhalf;
    float* o = out + (size_t)row0 * NC + k * D + m;
#pragma unroll
    for (int r = 0; r < 8; ++r) {
        if (row0 + r < N) {
            o[(size_t)r * NC +  0] = c0[r] + bv0;
            o[(size_t)r * NC + 16] = c1[r] + bv1;
            o[(size_t)r * NC + 32] = c2[r] + bv2;
            o[(size_t)r * NC + 48] = c3[r] + bv3;
        }
    }
}

// ---------------------------------------------------------------------------
extern "C" void kernel_launch(void* const* d_in, const int* in_sizes, int n_in,
                              void* d_out, int out_size, void* d_ws, size_t ws_size,
                              hipStream_t stream) {
    const float* x  = (const float*)d_in[0];   // (N, 64) f32
    const int*   ei = (const int*)d_in[1];     // (2, E)  int
    const float* ew = (const float*)d_in[2];   // (E,)    f32
    const float* W  = (const float*)d_in[3];   // (K, 64, 64) f32
    const float* b  = (const float*)d_in[4];   // (K, 64) f32
    float* out = (float*)d_out;                // (N, K, 64) f32

    const int N = in_sizes[0] / D;
    const int E = in_sizes[2];
    const int K = in_sizes[3] / (D * D);

    const int* src = ei;
    const int* dst = ei + E;

    float* ws   = (float*)d_ws;
    float* deg  = ws;               // N floats
    float* dinv = ws + N;           // N floats
    float* agg  = ws + 2 * N;       // N*64 floats

    int blk;
    blk = (N + 255) / 256;
    k_deg_init<<<blk, 256, 0, stream>>>(deg, N);

    blk = (E + 255) / 256;
    k_deg_edges<<<blk, 256, 0, stream>>>(dst, ew, deg, E);

    blk = (N + 255) / 256;
    k_dinv<<<blk, 256, 0, stream>>>(deg, dinv, N);

    blk = (N * (D / 4) + 255) / 256;
    k_agg_init<<<blk, 256, 0, stream>>>(x, dinv, agg, N);

    long scatter_threads = (long)E * 32;
    blk = (int)((scatter_threads + 255) / 256);
    k_scatter<<<blk, 256, 0, stream>>>(src, dst, ew, dinv, x, agg, E);

    int rowblks = (N + 15) / 16;               // 3125
    k_gemm<<<rowblks, 32 * K, 0, stream>>>(agg, W, b, out, N, K);
}